// Net_65695819760297
// MI455X (gfx1250) — compile-verified
//
#include <hip/hip_runtime.h>

typedef float v2f __attribute__((ext_vector_type(2)));
typedef float v8f __attribute__((ext_vector_type(8)));

#define FP_TOL 1e-3f
#define FP_MAX_ITERS 1000

// One wave32 handles 16 batch rows as the 16 columns of a
// V_WMMA_F32_16X16X4_F32:  D(16x16) = Wfc_pad(16x4) * U^T(4x16) + C.
// W_fc occupies A rows 0-2 AND 8-10, and the per-column injection
// c = b_fc + b_in + W_in*d occupies C rows 0-2 (lanes 0-15) AND 8-10
// (lanes 16-31). Result: each D VGPR j holds u'_j[n] replicated in both
// lane halves, so B's K=2 row (lanes 16-31 of VGPR0) is rebuilt with a
// single v_cndmask -- no cross-lane shuffle in the loop at all.
__global__ __launch_bounds__(256)
void deq_fixed_point_wmma(const float* __restrict__ d,
                          const float* __restrict__ Wfc,
                          const float* __restrict__ bfc,
                          const float* __restrict__ Win,
                          const float* __restrict__ bin,
                          const float* __restrict__ Wout,
                          const float* __restrict__ bout,
                          float* __restrict__ out,
                          int B)
{
    const int lane   = threadIdx.x & 31;
    const int warpId = (blockIdx.x * blockDim.x + threadIdx.x) >> 5;
    const int m      = lane & 15;          // A-matrix row id / batch column id
    const bool hi    = lane >= 16;         // K in {2,3} half of A/B layout

    const int row    = warpId * 16 + m;    // batch row owned by this column
    const int rowc   = (row < B) ? row : (B - 1);   // clamp, never mask EXEC
    const bool cvalid = (!hi) && (row < B);

    // ---- stream input row (coalesced; hi lanes duplicate their partner) ----
    const float* dp = d + (size_t)rowc * 3;
    const float d0 = dp[0], d1 = dp[1], d2 = dp[2];

    // ---- wave-uniform weights (scalar loads) ----
    const float w00 = Wfc[0], w01 = Wfc[1], w02 = Wfc[2];
    const float w10 = Wfc[3], w11 = Wfc[4], w12 = Wfc[5];
    const float w20 = Wfc[6], w21 = Wfc[7], w22 = Wfc[8];

    // ---- A = Wfc in rows {0,1,2} and {8,9,10}, zero elsewhere ----
    // A layout (f32 16x4): lane m holds row M=m (lanes 0-15) with K={0,1} in
    // VGPR{0,1}; lanes 16-31 hold the same rows with K={2,3}.
    float a0 = 0.f, a1 = 0.f;
    const int j = m & 7;                   // rows 0-2 and 8-10 -> j = 0..2
    if ((m & 7) < 3 && (m < 3 || m >= 8)) {
        if (j == 0)      { a0 = hi ? w02 : w00; a1 = hi ? 0.f : w01; }
        else if (j == 1) { a0 = hi ? w12 : w10; a1 = hi ? 0.f : w11; }
        else             { a0 = hi ? w22 : w20; a1 = hi ? 0.f : w21; }
    }
    v2f amat; amat.x = a0; amat.y = a1;

    // ---- injection folded into the WMMA C operand: c = b_fc + b_in + Win*d.
    // All 32 lanes keep their value: lanes 0-15 feed C rows 0-2, lanes 16-31
    // feed C rows 8-10 (same column, duplicated on purpose).
    const float bi0 = bfc[0] + bin[0];
    const float bi1 = bfc[1] + bin[1];
    const float bi2 = bfc[2] + bin[2];
    const float c0 = bi0 + Win[0]*d0 + Win[1]*d1 + Win[2]*d2;
    const float c1 = bi1 + Win[3]*d0 + Win[4]*d1 + Win[5]*d2;
    const float c2 = bi2 + Win[6]*d0 + Win[7]*d1 + Win[8]*d2;
    v8f cmat = {c0, c1, c2, 0.f, 0.f, 0.f, 0.f, 0.f};

    // ---- fixed-point iteration on the matrix pipe ----
    // B layout (f32 4x16): VGPR0 = row K(hi?2:0), VGPR1 = row K(hi?3:1);
    // column N = lane&15. K=3 row is don't-care (A column 3 is zero).
    v2f bmat; bmat.x = 0.f; bmat.y = 0.f;      // u0 = 0
    float u0 = 0.f, u1 = 0.f, u2 = 0.f;

    for (int it = 0; it < FP_MAX_ITERS; ++it) {
        // EXEC is all-1s here: no divergent branches are open in this loop.
        v8f acc = __builtin_amdgcn_wmma_f32_16x16x4_f32(
            /*neg_a=*/false, amat, /*neg_b=*/false, bmat,
            /*c_mod=*/(short)0, cmat, /*reuse_a=*/false, /*reuse_b=*/false);

        // u'_j, replicated in both lane halves (rows j and 8+j of D)
        const float n0 = fmaxf(acc[0], 0.f);
        const float n1 = fmaxf(acc[1], 0.f);
        const float n2 = fmaxf(acc[2], 0.f);

        const float r = fmaxf(fmaxf(fabsf(n0 - u0), fabsf(n1 - u1)),
                              fabsf(n2 - u2));
        u0 = n0; u1 = n1; u2 = n2;

        // rebuild B: K=2 row needs u'_2 in lanes 16-31 -- already there.
        bmat.x = hi ? n2 : n0;                 // single v_cndmask
        bmat.y = n1;

        if (__all(r < FP_TOL)) break;          // wave-uniform exit
    }

    // ---- output head (predicated store only, after all WMMAs) ----
    const float o = bout[0] + Wout[0]*u0 + Wout[1]*u1 + Wout[2]*u2;
    if (cvalid) out[row] = o;
}

extern "C" void kernel_launch(void* const* d_in, const int* in_sizes, int n_in,
                              void* d_out, int out_size, void* d_ws, size_t ws_size,
                              hipStream_t stream) {
    const float* d    = (const float*)d_in[0];
    const float* Wfc  = (const float*)d_in[1];
    const float* bfc  = (const float*)d_in[2];
    const float* Win  = (const float*)d_in[3];
    const float* bin  = (const float*)d_in[4];
    const float* Wout = (const float*)d_in[5];
    const float* bout = (const float*)d_in[6];
    float* out = (float*)d_out;

    const int B = in_sizes[0] / 3;             // [B,3] input
    const int rowsPerBlock = 128;              // 8 wave32 * 16 rows each
    const int blocks = (B + rowsPerBlock - 1) / rowsPerBlock;

    deq_fixed_point_wmma<<<blocks, 256, 0, stream>>>(
        d, Wfc, bfc, Win, bin, Wout, bout, out, B);
}